// WaveletFrontEnd_75531294867720
// MI455X (gfx1250) — compile-verified
//
#include <hip/hip_runtime.h>
#include <hip/hip_bf16.h>

typedef __attribute__((ext_vector_type(2))) float v2f;
typedef __attribute__((ext_vector_type(4))) float v4f;
typedef __attribute__((ext_vector_type(8))) float v8f;

#define NSC    256
#define LMAX   6145
#define TLEN   8192
#define NB     4
#define PAD    3072
#define PADT   14352          // 3072 + 8192 + 3072 + 16 slack (covers tap<=6151)

#define HROW   3076           // half-row stream: 2 floats * 1538 chunks (16B-mult)
#define BANKSZ (NSC * HROW)   // one half-bank (L or H) for one component
#define XROW   7176           // xs row: 4 floats * 1794 groups (16B-mult)

// ws layout (floats):
#define WS_MAX 0                      // 16 floats: per-batch max (uint bits)
#define WS_KRL 16                     // kR low-half  streams (taps 0,1 mod 4)
#define WS_KRH (WS_KRL + BANKSZ)      // kR high-half streams (taps 2,3 mod 4)
#define WS_KIL (WS_KRH + BANKSZ)
#define WS_KIH (WS_KIL + BANKSZ)
#define WS_XS  (WS_KIH + BANKSZ)      // 4 batches * 8 residues * XROW
#define WS_END (WS_XS + NB * 8 * XROW)

// ---------------- prep 1: split-repack bank into L/H per-lane streams ----------------
__global__ void wvl_prep_k(const float* __restrict__ kR, const float* __restrict__ kI,
                           float* __restrict__ ws) {
    int total = NSC * HROW;
    for (int i = blockIdx.x * blockDim.x + threadIdx.x; i < total;
         i += gridDim.x * blockDim.x) {
        int s   = i / HROW;
        int pos = i - s * HROW;          // 2*chunk + j
        int cc  = pos >> 1;
        int j   = pos & 1;
        int tapL = 4 * cc + j;           // lanes 0-15 K slots
        int tapH = 4 * cc + 2 + j;       // lanes 16-31 K slots
        const float* rR = kR + s * LMAX;
        const float* rI = kI + s * LMAX;
        ws[WS_KRL + i] = (tapL < LMAX) ? rR[tapL] : 0.0f;
        ws[WS_KRH + i] = (tapH < LMAX) ? rR[tapH] : 0.0f;
        ws[WS_KIL + i] = (tapL < LMAX) ? rI[tapL] : 0.0f;
        ws[WS_KIH + i] = (tapH < LMAX) ? rI[tapH] : 0.0f;
    }
}

// ---------------- prep 2: zero max slots + residue-compacted signal copies ----------------
// xs[b][r][4w+t] = x_padded(8w + r + {0,1,4,5}[t])
__global__ void wvl_prep_x(const float* __restrict__ x, float* __restrict__ ws) {
    if (blockIdx.x == 0 && threadIdx.x < 16)
        ((unsigned int*)ws)[threadIdx.x] = 0u;
    int total = NB * 8 * XROW;
    for (int i = blockIdx.x * blockDim.x + threadIdx.x; i < total;
         i += gridDim.x * blockDim.x) {
        int b   = i / (8 * XROW);
        int rem = i - b * (8 * XROW);
        int r   = rem / XROW;
        int k   = rem - r * XROW;
        int w   = k >> 2;
        int t   = k & 3;
        int dt  = (t & 1) + ((t >> 1) << 2);   // {0,1,4,5}
        int p   = 8 * w + r + dt;              // padded-domain index
        int src = p - PAD;
        ws[WS_XS + i] = (src >= 0 && src < TLEN) ? x[b * TLEN + src] : 0.0f;
    }
}

// ---------------- conv: 16 scales x 16 times per wave, f32 WMMA ----------------
__global__ __launch_bounds__(256) void wvl_conv_wmma(
    float* __restrict__ ws, float* __restrict__ out) {

    // grid decode: 16 scale-tiles x 64 time-groups x 4 batches
    unsigned bid = blockIdx.x;
    int st = bid & 15;
    int tg = (bid >> 4) & 63;
    int b  = bid >> 10;

    int wave = threadIdx.x >> 5;
    int lane = threadIdx.x & 31;
    int n  = lane & 15;
    int hi = lane >> 4;

    int s0 = st * 16;
    int t0 = (tg * 8 + wave) * 16;

    // per-tile K-start from analytic scale law: s_i = 2 * 256^(i/255)
    float smax = 2.0f * powf(256.0f, (float)(s0 + 15) * (1.0f / 255.0f));
    int len = 2 * (int)ceilf(6.0f * smax) + 1;
    int start_tap = LMAX - len - 8;
    int c0 = start_tap > 0 ? (start_tap >> 2) : 0;
    int c0e = c0 & ~1;                         // even chunk start

    // per-lane stream pointers (all 16B aligned at offsets 2c, c even)
    const float* kRp = ws + WS_KRL + (size_t)hi * (2 * BANKSZ - BANKSZ + BANKSZ) * 0
                     + (size_t)hi * BANKSZ + (size_t)(s0 + n) * HROW;
    const float* kIp = ws + WS_KIL + (size_t)hi * BANKSZ + (size_t)(s0 + n) * HROW;

    int Q  = t0 + n + 2 * hi;                  // padded-domain start index
    int r  = Q & 7;
    int Qd = Q >> 3;
    const float* xb = ws + WS_XS + (size_t)(b * 8 + r) * XROW + 4 * (size_t)Qd;

    v8f cR = {};
    v8f cI = {};

    // K loop: 2 chunks (8 taps) per iteration; streams zero-tailed past LMAX
#pragma unroll 2
    for (int c = c0e; c <= 1536; c += 2) {
        int o = 2 * c;
        v4f aR = *reinterpret_cast<const v4f*>(kRp + o);
        v4f aI = *reinterpret_cast<const v4f*>(kIp + o);
        v4f bx = *reinterpret_cast<const v4f*>(xb + o);
        v2f ar0 = {aR.x, aR.y}, ar1 = {aR.z, aR.w};
        v2f ai0 = {aI.x, aI.y}, ai1 = {aI.z, aI.w};
        v2f bx0 = {bx.x, bx.y}, bx1 = {bx.z, bx.w};
        cR = __builtin_amdgcn_wmma_f32_16x16x4_f32(false, ar0, false, bx0,
                                                   (short)0, cR, false, false);
        cI = __builtin_amdgcn_wmma_f32_16x16x4_f32(false, ai0, false, bx0,
                                                   (short)0, cI, false, false);
        cR = __builtin_amdgcn_wmma_f32_16x16x4_f32(false, ar1, false, bx1,
                                                   (short)0, cR, false, false);
        cI = __builtin_amdgcn_wmma_f32_16x16x4_f32(false, ai1, false, bx1,
                                                   (short)0, cI, false, false);
    }

    // write interleaved complex output + local magnitude max
    float m = 1e-8f;
#pragma unroll
    for (int v = 0; v < 8; ++v) {
        float yr = cR[v];
        float yi = cI[v];
        size_t row = (size_t)(b * NSC + s0 + 8 * hi + v);
        size_t idx = row * (size_t)TLEN + (size_t)(t0 + n);
        float2 p;
        p.x = yr; p.y = yi;
        reinterpret_cast<float2*>(out)[idx] = p;
        m = fmaxf(m, sqrtf(yr * yr + yi * yi));
    }

    // wave-level max reduce, then one atomic per wave (bit-compare valid: m > 0)
    for (int o = 16; o > 0; o >>= 1)
        m = fmaxf(m, __shfl_xor(m, o, 32));
    if (lane == 0)
        atomicMax((unsigned int*)ws + b, __float_as_uint(m));
}

// ---------------- mag_db = 20*log10(mag / denom) ----------------
__global__ void wvl_magdb(const float* __restrict__ ws, float* __restrict__ out) {
    const unsigned int* wsmax = (const unsigned int*)ws;
    const size_t YTOT = (size_t)NB * NSC * TLEN;   // 8388608
    for (size_t i = (size_t)blockIdx.x * blockDim.x + threadIdx.x; i < YTOT;
         i += (size_t)gridDim.x * blockDim.x) {
        int b = (int)(i / ((size_t)NSC * TLEN));
        float yr = out[2 * i];
        float yi = out[2 * i + 1];
        float mag = fmaxf(sqrtf(yr * yr + yi * yi), 1e-8f);
        float denom = fmaxf(__uint_as_float(wsmax[b]), 1e-8f);
        out[2 * YTOT + i] = 20.0f * log10f(mag / denom);
    }
}

extern "C" void kernel_launch(void* const* d_in, const int* in_sizes, int n_in,
                              void* d_out, int out_size, void* d_ws, size_t ws_size,
                              hipStream_t stream) {
    const float* x  = (const float*)d_in[0];
    const float* kR = (const float*)d_in[1];
    const float* kI = (const float*)d_in[2];
    float* out = (float*)d_out;
    float* ws  = (float*)d_ws;

    // 1) split-repack bank + residue-compacted signal copies
    wvl_prep_k<<<2048, 256, 0, stream>>>(kR, kI, ws);
    wvl_prep_x<<<256, 256, 0, stream>>>(x, ws);

    // 2) WMMA conv: 16 scale-tiles * 64 time-groups * 4 batches = 4096 blocks
    wvl_conv_wmma<<<16 * 64 * NB, 256, 0, stream>>>(ws, out);

    // 3) dB magnitude
    wvl_magdb<<<8192, 256, 0, stream>>>(ws, out);
}